// MVDR_Beamforming_55748675502292
// MI455X (gfx1250) — compile-verified
//
#include <hip/hip_runtime.h>
#include <math.h>

// Problem dims (from reference setup_inputs)
#define B_  16
#define M_  6
#define F_  513
#define T_  1500
#define NBF (B_ * F_)   // 8208

typedef __attribute__((ext_vector_type(2))) float v2f;
typedef __attribute__((ext_vector_type(8))) float v8f;

// ---------------------------------------------------------------------------
// Kernel 1: per-(b,f) covariance via V_WMMA_F32_16X16X4_F32.
// One wave per (b,f). A fragment rows 0..5 = y = x*mask, rows 6..11 = x,
// rows 12..15 = 0.  D = A * A^T  ->  D[0:6,0:6] = sum_t y y^T (= Phiss*Msum),
// D[6:12,6:12] = sum_t x x^T (= Phixx*T).  Per the ISA VGPR layouts the A
// fragment and the B(=A^T) fragment are identical register contents, so one
// register pair feeds both WMMA sources.
//
// Branchless inner loop: every lane loads from a valid (clamped) row; the
// row-type predicates are folded into lane-constant selects so the loop body
// is 2 loads + a few VALU + 1 WMMA with no exec-mask manipulation.
// ---------------------------------------------------------------------------
__global__ __launch_bounds__(256) void cov_kernel(const float* __restrict__ x,
                                                  const float* __restrict__ mask,
                                                  float* __restrict__ phiss,   // [NBF][36]
                                                  float* __restrict__ phixx)   // [NBF][36]
{
    const int wid  = (blockIdx.x * blockDim.x + threadIdx.x) >> 5;  // wave id = (b,f) id
    if (wid >= NBF) return;                                          // wave-uniform
    const int lane = threadIdx.x & 31;
    const int b = wid / F_;
    const int f = wid % F_;

    // A-fragment lane mapping (32-bit A 16x4): row = lane%16,
    // VGPR0/1 hold K = {0,1} for lanes 0..15 and K = {2,3} for lanes 16..31.
    const int  r     = lane & 15;
    const int  lo    = (lane >> 4) * 2;           // K offset inside the 4-wide chunk
    const bool isY   = (r < 6);
    const bool isRow = (r < 12);
    const int  m     = isY ? r : (isRow ? (r - 6) : 0);   // clamped: always valid row

    const float selY   = isY   ? 1.0f : 0.0f;     // contributes to msum
    const float selRow = isRow ? 1.0f : 0.0f;     // x-row passthrough vs dead row

    const size_t rowoff = ((size_t)(b * M_ + m) * F_ + f) * (size_t)T_;
    const float* xp = x    + rowoff + lo;         // per-lane base, advance by t0
    const float* mp = mask + rowoff + lo;

    v8f   acc0 = {};
    v8f   acc1 = {};
    float msum = 0.0f;   // sum over (mic,t) of mask^2 (lanes r<6 in both halves cover all K)

    auto frag = [&](int t0) -> v2f {
        const float2 vx = *reinterpret_cast<const float2*>(xp + t0);
        const float2 vm = *reinterpret_cast<const float2*>(mp + t0);
        msum = fmaf(vm.x * vm.x + vm.y * vm.y, selY, msum);
        // y rows: multiply by mask; x rows: multiply by 1; dead rows: by 0.
        const float qx = isY ? vm.x : selRow;     // lane-constant cond -> v_cndmask
        const float qy = isY ? vm.y : selRow;
        v2f a;
        a.x = vx.x * qx;
        a.y = vx.y * qy;
        return a;
    };

    int t0 = 0;
    // Two independent accumulator chains for ILP (374 chunks paired), tail chunk after.
    for (; t0 + 8 <= T_; t0 += 8) {
        v2f a0 = frag(t0);
        v2f a1 = frag(t0 + 4);
        acc0 = __builtin_amdgcn_wmma_f32_16x16x4_f32(false, a0, false, a0, (short)0, acc0, false, false);
        acc1 = __builtin_amdgcn_wmma_f32_16x16x4_f32(false, a1, false, a1, (short)0, acc1, false, false);
    }
    for (; t0 < T_; t0 += 4) {
        v2f a0 = frag(t0);
        acc0 = __builtin_amdgcn_wmma_f32_16x16x4_f32(false, a0, false, a0, (short)0, acc0, false, false);
    }
    v8f acc = acc0 + acc1;

    // Wave-wide reduction of msum (wave32).
    for (int off = 16; off > 0; off >>= 1)
        msum += __shfl_xor(msum, off, 32);

    // C/D layout: D[mm][nn] -> lane = nn + (mm>=8 ? 16 : 0), vgpr = mm & 7.
    const float invms = 1.0f / msum;
    const float invT  = 1.0f / (float)T_;
    float* ps = phiss + (size_t)wid * 36;
    float* px = phixx + (size_t)wid * 36;

    if (lane < 6) {                      // Phiss[i][j], j = lane, i = vgpr 0..5
        const int j = lane;
        #pragma unroll
        for (int i = 0; i < 6; ++i) ps[i * 6 + j] = acc[i] * invms;
    } else if (lane < 12) {              // Phixx rows 0,1 (D rows 6,7), j = lane-6
        const int j = lane - 6;
        px[0 * 6 + j] = acc[6] * invT;
        px[1 * 6 + j] = acc[7] * invT;
    } else if (lane >= 22 && lane < 28) {// Phixx rows 2..5 (D rows 8..11), j = lane-22
        const int j = lane - 22;
        #pragma unroll
        for (int v = 0; v < 4; ++v) px[(v + 2) * 6 + j] = acc[v] * invT;
    }
}

// ---------------------------------------------------------------------------
// Kernel 2: per-(b,f) small solve. One thread each:
//   Phinn = Phixx - Phiss ; inv(Phinn) by Gauss-Jordan w/ pivoting
//   s = singular values of (Phiss - colmean) via Jacobi on C^T C, sorted desc
//   w = Phinn^{-1} s / (s^T Phinn^{-1} s)
// ---------------------------------------------------------------------------
__global__ __launch_bounds__(256) void solve_kernel(const float* __restrict__ phiss,
                                                    const float* __restrict__ phixx,
                                                    float* __restrict__ wout)   // [NBF][6]
{
    const int tid = blockIdx.x * blockDim.x + threadIdx.x;
    if (tid >= NBF) return;

    float ps[6][6], nn[6][6];
    const float* p0 = phiss + (size_t)tid * 36;
    const float* p1 = phixx + (size_t)tid * 36;
    for (int i = 0; i < 6; ++i)
        for (int j = 0; j < 6; ++j) {
            const float s = p0[i * 6 + j];
            ps[i][j] = s;
            nn[i][j] = p1[i * 6 + j] - s;
        }

    // centered = ps - mean over rows (per column); G = centered^T centered
    float cen[6][6], G[6][6];
    for (int j = 0; j < 6; ++j) {
        float mcol = 0.0f;
        for (int i = 0; i < 6; ++i) mcol += ps[i][j];
        mcol *= (1.0f / 6.0f);
        for (int i = 0; i < 6; ++i) cen[i][j] = ps[i][j] - mcol;
    }
    for (int a = 0; a < 6; ++a)
        for (int c = 0; c < 6; ++c) {
            float s = 0.0f;
            for (int i = 0; i < 6; ++i) s += cen[i][a] * cen[i][c];
            G[a][c] = s;
        }

    // Cyclic Jacobi eigenvalue sweeps on symmetric G.
    for (int sweep = 0; sweep < 10; ++sweep) {
        for (int p = 0; p < 5; ++p)
            for (int q = p + 1; q < 6; ++q) {
                const float apq = G[p][q];
                if (fabsf(apq) < 1e-30f) continue;
                const float tau = (G[q][q] - G[p][p]) / (2.0f * apq);
                const float t   = (tau >= 0.0f ? 1.0f : -1.0f) /
                                  (fabsf(tau) + sqrtf(1.0f + tau * tau));
                const float c   = rsqrtf(1.0f + t * t);
                const float s   = t * c;
                for (int k = 0; k < 6; ++k) {            // G <- G * J
                    const float gkp = G[k][p], gkq = G[k][q];
                    G[k][p] = c * gkp - s * gkq;
                    G[k][q] = s * gkp + c * gkq;
                }
                for (int k = 0; k < 6; ++k) {            // G <- J^T * G
                    const float gpk = G[p][k], gqk = G[q][k];
                    G[p][k] = c * gpk - s * gqk;
                    G[q][k] = s * gpk + c * gqk;
                }
            }
    }
    float sv[6];
    for (int i = 0; i < 6; ++i) sv[i] = sqrtf(fmaxf(G[i][i], 0.0f));
    for (int i = 0; i < 5; ++i)                            // sort descending (svd order)
        for (int j = i + 1; j < 6; ++j)
            if (sv[j] > sv[i]) { const float t = sv[i]; sv[i] = sv[j]; sv[j] = t; }

    // Gauss-Jordan inverse of nn with partial pivoting.
    float inv[6][6];
    for (int i = 0; i < 6; ++i)
        for (int j = 0; j < 6; ++j) inv[i][j] = (i == j) ? 1.0f : 0.0f;
    for (int k = 0; k < 6; ++k) {
        int piv = k; float big = fabsf(nn[k][k]);
        for (int rr = k + 1; rr < 6; ++rr) {
            const float v = fabsf(nn[rr][k]);
            if (v > big) { big = v; piv = rr; }
        }
        if (piv != k)
            for (int j = 0; j < 6; ++j) {
                float t = nn[k][j];  nn[k][j]  = nn[piv][j];  nn[piv][j]  = t;
                t = inv[k][j];       inv[k][j] = inv[piv][j]; inv[piv][j] = t;
            }
        const float pivinv = 1.0f / nn[k][k];
        for (int j = 0; j < 6; ++j) { nn[k][j] *= pivinv; inv[k][j] *= pivinv; }
        for (int rr = 0; rr < 6; ++rr) {
            if (rr == k) continue;
            const float fs = nn[rr][k];
            for (int j = 0; j < 6; ++j) {
                nn[rr][j]  -= fs * nn[k][j];
                inv[rr][j] -= fs * inv[k][j];
            }
        }
    }

    float num[6], den = 0.0f;
    for (int i = 0; i < 6; ++i) {
        float s = 0.0f;
        for (int j = 0; j < 6; ++j) s += inv[i][j] * sv[j];
        num[i] = s;
        den += sv[i] * s;
    }
    const float dinv = 1.0f / den;
    float* wo = wout + (size_t)tid * 6;
    for (int i = 0; i < 6; ++i) wo[i] = num[i] * dinv;
}

// ---------------------------------------------------------------------------
// Kernel 3: beamform every frame, float4-vectorized (bandwidth-bound pass).
// out[b,f,t] = sum_m w[b,f,m] * x[b,m,f,t] * mask[b,m,f,t]
// ---------------------------------------------------------------------------
__global__ __launch_bounds__(256) void beamform_kernel(const float* __restrict__ x,
                                                       const float* __restrict__ mask,
                                                       const float* __restrict__ wv,
                                                       float* __restrict__ out)
{
    const int T4  = T_ / 4;                 // 375, T_ % 4 == 0 so all float4 aligned
    const int idx = blockIdx.x * blockDim.x + threadIdx.x;
    if (idx >= NBF * T4) return;
    const int wid = idx / T4;
    const int t   = (idx - wid * T4) * 4;
    const int b   = wid / F_;
    const int f   = wid % F_;

    const float* wp = wv + (size_t)wid * 6;
    float4 a; a.x = 0.0f; a.y = 0.0f; a.z = 0.0f; a.w = 0.0f;
    #pragma unroll
    for (int m = 0; m < 6; ++m) {
        const size_t off = ((size_t)(b * M_ + m) * F_ + f) * (size_t)T_ + t;
        const float4 vx = *reinterpret_cast<const float4*>(x + off);
        const float4 vm = *reinterpret_cast<const float4*>(mask + off);
        const float  wm = wp[m];
        a.x += wm * vx.x * vm.x;
        a.y += wm * vx.y * vm.y;
        a.z += wm * vx.z * vm.z;
        a.w += wm * vx.w * vm.w;
    }
    *reinterpret_cast<float4*>(out + (size_t)wid * T_ + t) = a;
}

// ---------------------------------------------------------------------------
extern "C" void kernel_launch(void* const* d_in, const int* in_sizes, int n_in,
                              void* d_out, int out_size, void* d_ws, size_t ws_size,
                              hipStream_t stream) {
    (void)in_sizes; (void)n_in; (void)out_size; (void)ws_size;
    const float* x    = (const float*)d_in[0];
    const float* mask = (const float*)d_in[1];
    float* out = (float*)d_out;

    // Workspace layout: Phiss[NBF][36] | Phixx[NBF][36] | w[NBF][6]  (~2.6 MB)
    float* phiss = (float*)d_ws;
    float* phixx = phiss + (size_t)NBF * 36;
    float* wv    = phixx + (size_t)NBF * 36;

    // Phase 1: one wave per (b,f): 8208 waves, 8 waves (256 thr) per block.
    const int blocks1 = (NBF * 32 + 255) / 256;   // 1026
    hipLaunchKernelGGL(cov_kernel, dim3(blocks1), dim3(256), 0, stream,
                       x, mask, phiss, phixx);

    // Phase 2: one thread per (b,f).
    const int blocks2 = (NBF + 255) / 256;
    hipLaunchKernelGGL(solve_kernel, dim3(blocks2), dim3(256), 0, stream,
                       phiss, phixx, wv);

    // Phase 3: one thread per 4 frames.
    const int total3  = NBF * (T_ / 4);
    const int blocks3 = (total3 + 255) / 256;
    hipLaunchKernelGGL(beamform_kernel, dim3(blocks3), dim3(256), 0, stream,
                       x, mask, wv, out);
}